// TypedCoords2Volume_8521215115551
// MI455X (gfx1250) — compile-verified
//
#include <hip/hip_runtime.h>

// Problem constants from the reference
constexpr int kBox    = 120;
constexpr int kNTypes = 11;
constexpr int kCap    = 372;
constexpr int kAtoms  = kNTypes * kCap;           // 4092 atoms per sample
constexpr long long kVol = (long long)kBox * kBox * kBox;

// ---------------------------------------------------------------------------
// Kernel 1: zero-fill d_out (608 MB) via the CDNA5 async LDS->global store
// engine. Each thread owns a 16B zero slab in LDS; the loop only produces
// addresses while global_store_async_from_lds_b128 streams zeros to HBM.
// Tracked by ASYNCcnt; drained with s_wait_asynccnt before wave exit.
// ---------------------------------------------------------------------------
__global__ void zero_fill_async(float* __restrict__ out, long long n4, long long n) {
    __shared__ __align__(16) float zslab[256 * 4];     // 16 B per thread
    float4* zs = (float4*)zslab;
    zs[threadIdx.x] = make_float4(0.f, 0.f, 0.f, 0.f);
    __syncthreads();

    // Low 32 bits of a generic pointer into LDS == LDS byte offset (aperture form)
    unsigned lds_off = (unsigned)(size_t)&zs[threadIdx.x];

    float4* o4 = (float4*)out;
    long long stride = (long long)gridDim.x * blockDim.x;
    for (long long i = (long long)blockIdx.x * blockDim.x + threadIdx.x;
         i < n4; i += stride) {
        float4* p = o4 + i;
        asm volatile("global_store_async_from_lds_b128 %0, %1, off"
                     :: "v"(p), "v"(lds_off)
                     : "memory");
    }

    // Scalar tail (out_size % 4 elements), handled by block 0
    long long tail = n - (n4 << 2);
    if (blockIdx.x == 0 && (long long)threadIdx.x < tail) {
        out[(n4 << 2) + threadIdx.x] = 0.f;
    }

    asm volatile("s_wait_asynccnt 0" ::: "memory");
}

// ---------------------------------------------------------------------------
// Kernel 2: trilinear splat. One thread per atom; 8 corner atomics.
// ---------------------------------------------------------------------------
__global__ void splat_kernel(const float* __restrict__ coords,
                             const int* __restrict__ num_atoms,
                             const int* __restrict__ offsets,
                             float* __restrict__ out, int nB) {
    int g = blockIdx.x * blockDim.x + threadIdx.x;
    int total = nB * kAtoms;
    if (g >= total) return;

    int b = g / kAtoms;
    int a = g - b * kAtoms;

    const float* c = coords + (size_t)b * (3 * kAtoms) + 3 * (size_t)a;
    float x = c[0], y = c[1], z = c[2];            // RES == 1.0

    // Type of this atom: first t with offsets[t] <= a < offsets[t]+num[t]
    const int* offs = offsets   + b * kNTypes;
    const int* nums = num_atoms + b * kNTypes;
    int atype = -1;
#pragma unroll
    for (int t = 0; t < kNTypes; ++t) {
        int o = offs[t];
        if (atype < 0 && a >= o && a < o + nums[t]) atype = t;
    }
    if (atype < 0) return;                          // not a valid atom slot

    float fx0 = floorf(x), fy0 = floorf(y), fz0 = floorf(z);
    int ix = (int)fx0, iy = (int)fy0, iz = (int)fz0;
    float fx = x - fx0, fy = y - fy0, fz = z - fz0;
    float wx[2] = {1.f - fx, fx};
    float wy[2] = {1.f - fy, fy};
    float wz[2] = {1.f - fz, fz};

    float* vol = out + ((size_t)b * kNTypes + atype) * (size_t)kVol;

#pragma unroll
    for (int k = 0; k < 8; ++k) {
        int dx = (k >> 2) & 1, dy = (k >> 1) & 1, dz = k & 1;
        int cx = ix + dx, cy = iy + dy, cz = iz + dz;
        if (cx >= 0 && cx < kBox && cy >= 0 && cy < kBox && cz >= 0 && cz < kBox) {
            float w = wx[dx] * wy[dy] * wz[dz];
            atomicAdd(&vol[((size_t)cx * kBox + cy) * kBox + cz], w);
        }
    }
}

// ---------------------------------------------------------------------------
extern "C" void kernel_launch(void* const* d_in, const int* in_sizes, int n_in,
                              void* d_out, int out_size, void* d_ws, size_t ws_size,
                              hipStream_t stream) {
    const float* coords    = (const float*)d_in[0];   // [B, 3A] f32
    const int*   num_atoms = (const int*)  d_in[1];   // [B, NTYPES] i32
    const int*   offsets   = (const int*)  d_in[2];   // [B, NTYPES] i32
    float*       out       = (float*)d_out;           // [B, NTYPES, BOX^3] f32

    long long n  = (long long)out_size;               // 152,064,000 floats
    long long n4 = n >> 2;                            // float4 stores

    // Zero the volume: bandwidth-bound, async LDS->global store path.
    zero_fill_async<<<4096, 256, 0, stream>>>(out, n4, n);

    // Scatter atoms (same stream -> ordered after the fill).
    int nB    = in_sizes[1] / kNTypes;
    int total = nB * kAtoms;
    splat_kernel<<<(total + 255) / 256, 256, 0, stream>>>(coords, num_atoms,
                                                          offsets, out, nB);
}